// BeamPPTreeEnsemble_28200755265905
// MI455X (gfx1250) — compile-verified
//
#include <hip/hip_runtime.h>
#include <hip/hip_bf16.h>

// Problem constants (match reference)
#define B_      32768
#define T_      512
#define D_      8
#define F_      512
#define PAD_    (T_ * 128)    // 65536 = T * 2^(D-1)
#define LEVELS_ (D_ - 1)      // 7
#define ROWS_   16            // samples staged per block (16*512*4 = 32KB LDS)

// Pointer types matching __builtin_amdgcn_global_load_async_to_lds_b128's
// signature: (v4i addrspace(1)*, v4i addrspace(3)*, Ii offset, Ii cpol)
typedef int v4i __attribute__((ext_vector_type(4)));
typedef __attribute__((address_space(1))) v4i g_v4i;   // global int4
typedef __attribute__((address_space(3))) v4i l_v4i;   // LDS int4

__global__ __launch_bounds__(T_) void gbdt_traverse_kernel(
    const float* __restrict__ x,            // (B, F)
    const int*   __restrict__ root_nodes,   // (T,)
    const float* __restrict__ root_biases,  // (T,)
    const int*   __restrict__ tree_indices, // (T,)  == 2*t
    const int*   __restrict__ nodes_lv,     // (LEVELS, PAD)
    const float* __restrict__ biases_lv,    // (LEVELS, PAD)
    const float* __restrict__ leaf_nodes,   // (T*2^D, 1)
    float*       __restrict__ out)          // (B, T)
{
    __shared__ float tileX[ROWS_ * F_];

    const int t    = threadIdx.x;           // one thread per tree
    const int row0 = blockIdx.x * ROWS_;

    // ---------------------------------------------------------------
    // Stage the 16-sample feature tile into LDS with gfx1250 async
    // global->LDS DMA (tracked by ASYNCcnt). 32KB tile, 16B per lane
    // per issue, 512 lanes -> 4 issues per thread.
    // ---------------------------------------------------------------
    {
        char* gbase = (char*)(x + (size_t)row0 * F_);   // tile base in global
        char* lbase = (char*)tileX;
        constexpr int CHUNKS = (ROWS_ * F_ * 4) / (16 * T_); // = 4
        #pragma unroll
        for (int k = 0; k < CHUNKS; ++k) {
            const int byteoff = (t + k * T_) * 16;
            __builtin_amdgcn_global_load_async_to_lds_b128(
                (g_v4i*)(gbase + byteoff),
                (l_v4i*)(lbase + byteoff),
                /*offset=*/0, /*cpol=*/0);
        }
    }

    // Overlap with the async copy in flight: per-tree root metadata
    // (these come from tiny L2-resident tables).
    const int   rf    = root_nodes[t];
    const float rb    = root_biases[t];
    const int   tbase = tree_indices[t];    // == 2*t

    // Wait for this wave's async DMA, then make the tile visible to all waves.
#if defined(__has_builtin) && __has_builtin(__builtin_amdgcn_s_wait_asynccnt)
    __builtin_amdgcn_s_wait_asynccnt(0);
#else
    asm volatile("s_wait_asynccnt 0" ::: "memory");
#endif
    __syncthreads();

    // ---------------------------------------------------------------
    // Walk all 8 levels for 4 samples at a time: 4 independent
    // dependency chains per thread so the scheduler can overlap the
    // L2 table gathers and LDS feature gathers (ds_load_b32) of all
    // four walks instead of serializing one 7-deep chain.
    // Node-table gathers are 4B scattered reads that stay resident in
    // the 192MB L2 (tables total ~4MB).
    // ---------------------------------------------------------------
    constexpr int SB = 4;                   // samples walked concurrently
    for (int s0 = 0; s0 < ROWS_; s0 += SB) {
        int prev[SB];

        #pragma unroll
        for (int j = 0; j < SB; ++j) {
            const float* xr = &tileX[(s0 + j) * F_];
            prev[j] = tbase + ((xr[rf] >= rb) ? 1 : 0);
        }

        #pragma unroll
        for (int l = 0; l < LEVELS_; ++l) {
            int   fi[SB];
            float bv[SB];
            #pragma unroll
            for (int j = 0; j < SB; ++j) {
                fi[j] = nodes_lv[l * PAD_ + prev[j]];
                bv[j] = biases_lv[l * PAD_ + prev[j]];
            }
            #pragma unroll
            for (int j = 0; j < SB; ++j) {
                const float* xr = &tileX[(s0 + j) * F_];
                const int bit = (xr[fi[j]] >= bv[j]) ? 1 : 0;
                prev[j] = 2 * prev[j] + bit;
            }
        }

        // Coalesced, streaming (non-temporal) stores: out[(row0+s)*T + t]
        #pragma unroll
        for (int j = 0; j < SB; ++j) {
            const float leaf = leaf_nodes[prev[j]];
            __builtin_nontemporal_store(leaf, &out[(size_t)(row0 + s0 + j) * T_ + t]);
        }
    }
}

extern "C" void kernel_launch(void* const* d_in, const int* in_sizes, int n_in,
                              void* d_out, int out_size, void* d_ws, size_t ws_size,
                              hipStream_t stream) {
    const float* x            = (const float*)d_in[0];
    const int*   root_nodes   = (const int*)  d_in[1];
    const float* root_biases  = (const float*)d_in[2];
    const int*   tree_indices = (const int*)  d_in[3];
    const int*   nodes_lv     = (const int*)  d_in[4];
    const float* biases_lv    = (const float*)d_in[5];
    const float* leaf_nodes   = (const float*)d_in[6];
    float*       out          = (float*)d_out;

    const int blocks = B_ / ROWS_;   // 2048 blocks, 512 threads (16 wave32s) each
    gbdt_traverse_kernel<<<blocks, T_, 0, stream>>>(
        x, root_nodes, root_biases, tree_indices, nodes_lv, biases_lv, leaf_nodes, out);
}